// GCN_66915590472494
// MI455X (gfx1250) — compile-verified
//
#include <hip/hip_runtime.h>
#include <hip/hip_bf16.h>

#define FEAT 96
#define XS_LD 97  // padded LDS stride to spread banks

typedef float v2f __attribute__((ext_vector_type(2)));
typedef float v8f __attribute__((ext_vector_type(8)));

// ---------------------------------------------------------------------------
// Degree / normalization precompute
// ---------------------------------------------------------------------------
__global__ void gcn_deg_init(float* __restrict__ deg, int n) {
    int i = blockIdx.x * blockDim.x + threadIdx.x;
    if (i < n) deg[i] = 1.0f;  // self-loop weight
}

__global__ void gcn_deg_accum(const int* __restrict__ dst,
                              const float* __restrict__ ew,
                              float* __restrict__ deg, int ne) {
    int e = blockIdx.x * blockDim.x + threadIdx.x;
    if (e < ne) atomicAdd(&deg[dst[e]], ew[e]);
}

__global__ void gcn_deg_finalize(float* __restrict__ deg, int n) {
    int i = blockIdx.x * blockDim.x + threadIdx.x;
    if (i < n) {
        float v = deg[i];
        deg[i] = (v > 0.0f) ? rsqrtf(v) : 0.0f;  // deg -> dis in place
    }
}

// ---------------------------------------------------------------------------
// Node-wise GEMM via fp32 WMMA with fused self-loop+bias epilogue:
//   H[r,c]   = sum_k act(X[r,k]) * W[k,c]
//   OUT[r,c] = dis[r]^2 * H[r,c] + bias[c]     (aggregation init)
//
// Block = 192 threads = 6 waves; block owns one 16-row tile, wave w owns
// column tile w (16 cols). X tile staged in LDS.
//
// V_WMMA_F32_16X16X4_F32 fragment layout (ISA 7.12.2, wave32):
//   A 16x4 : lane%16 = row M; lanes<16 hold K={k,k+1}, lanes>=16 K={k+2,k+3}
//   B 4x16 : lane%16 = col N; mirrored K split
//   C/D    : 8 VGPRs; N = lane%16, M = vgpr + 8*(lane>=16)
// ---------------------------------------------------------------------------
__global__ __launch_bounds__(192) void gcn_gemm_wmma(
    const float* __restrict__ X, const float* __restrict__ W,
    const float* __restrict__ dis, const float* __restrict__ bias,
    float* __restrict__ H, float* __restrict__ OUT,
    int nrows, int relu_in)
{
    __shared__ float xs[16 * XS_LD];

    const int tid  = threadIdx.x;
    const int row0 = blockIdx.x * 16;
    const bool full_tile = (row0 + 16 <= nrows);

    // cooperative, coalesced stage of the 16x96 X tile (ReLU fused)
    for (int i = tid; i < 16 * FEAT; i += 192) {
        int r = i / FEAT, c = i - r * FEAT;
        float v = 0.0f;
        if (full_tile || row0 + r < nrows) v = X[(size_t)(row0 + r) * FEAT + c];
        if (relu_in) v = v > 0.0f ? v : 0.0f;
        xs[r * XS_LD + c] = v;
    }
    __syncthreads();

    const int wave  = tid >> 5;        // 0..5 -> column tile
    const int lane  = tid & 31;
    const int col0  = wave * 16;
    const int m     = lane & 15;       // A: row-in-tile, B/D: col-in-tile
    const int khalf = (lane >> 4) * 2; // 0 or 2

    v8f acc = {0.f, 0.f, 0.f, 0.f, 0.f, 0.f, 0.f, 0.f};

    #pragma unroll 8
    for (int k = 0; k < FEAT; k += 4) {
        v2f a, b;
        a.x = xs[m * XS_LD + k + khalf + 0];
        a.y = xs[m * XS_LD + k + khalf + 1];
        b.x = W[(size_t)(k + khalf + 0) * FEAT + col0 + m];
        b.y = W[(size_t)(k + khalf + 1) * FEAT + col0 + m];
        acc = __builtin_amdgcn_wmma_f32_16x16x4_f32(
            /*neg_a=*/false, a, /*neg_b=*/false, b,
            /*c_mod=*/(short)0, acc, /*reuse_a=*/false, /*reuse_b=*/false);
    }

    // Epilogue: rows r = row0 + rbase + v, col = col0 + m
    const int rbase = (lane >> 4) * 8;
    const float bcol = bias[col0 + m];
    const size_t base = (size_t)(row0 + rbase) * FEAT + col0 + m;
    float* __restrict__ Hp = H + base;
    float* __restrict__ Op = OUT + base;
    const float* __restrict__ dp = dis + row0 + rbase;

    if (full_tile) {
        #pragma unroll
        for (int v = 0; v < 8; ++v) {
            float h = acc[v];
            float s = dp[v];          // half-wave-uniform broadcast load
            Hp[(size_t)v * FEAT] = h;
            Op[(size_t)v * FEAT] = s * s * h + bcol;
        }
    } else {
        #pragma unroll
        for (int v = 0; v < 8; ++v) {
            if (row0 + rbase + v < nrows) {
                float h = acc[v];
                float s = dp[v];
                Hp[(size_t)v * FEAT] = h;
                Op[(size_t)v * FEAT] = s * s * h + bcol;
            }
        }
    }
}

// ---------------------------------------------------------------------------
// Edge scatter: 24 threads per edge, each moves one float4 (contiguous 384B
// gather from H[src], 4x global_atomic_add_f32 into out[dst]). Feature arrays
// (19.2 MB) are L2-resident on the 192 MB L2, so this runs at L2 bandwidth.
// ---------------------------------------------------------------------------
__global__ void gcn_edge_scatter(const int* __restrict__ src,
                                 const int* __restrict__ dst,
                                 const float* __restrict__ ew,
                                 const float* __restrict__ dis,
                                 const float* __restrict__ H,
                                 float* __restrict__ out, int ne) {
    int idx = blockIdx.x * blockDim.x + threadIdx.x;
    int e = idx / 24;
    int c = idx - e * 24;
    if (e >= ne) return;
    int s = src[e], d = dst[e];
    float norm = dis[s] * ew[e] * dis[d];
    const float4 v = ((const float4*)(H + (size_t)s * FEAT))[c];
    float* o = out + (size_t)d * FEAT + c * 4;
    atomicAdd(o + 0, norm * v.x);
    atomicAdd(o + 1, norm * v.y);
    atomicAdd(o + 2, norm * v.z);
    atomicAdd(o + 3, norm * v.w);
}

// ---------------------------------------------------------------------------
// Launch: deg/dis once, then 3x (fused GEMM+init -> edge scatter),
// ping-ponging two N*96 buffers in workspace. All stream-ordered.
// ---------------------------------------------------------------------------
extern "C" void kernel_launch(void* const* d_in, const int* in_sizes, int n_in,
                              void* d_out, int out_size, void* d_ws, size_t ws_size,
                              hipStream_t stream) {
    const float* x  = (const float*)d_in[0];
    const int*   ei = (const int*)  d_in[1];
    const float* ew = (const float*)d_in[2];
    const float* W1 = (const float*)d_in[3];
    const float* b1 = (const float*)d_in[4];
    const float* W2 = (const float*)d_in[5];
    const float* b2 = (const float*)d_in[6];
    const float* W3 = (const float*)d_in[7];
    const float* b3 = (const float*)d_in[8];

    const int N = in_sizes[0] / FEAT;
    const int E = in_sizes[2];
    const int* src = ei;
    const int* dst = ei + E;

    float* ws  = (float*)d_ws;
    float* dis = ws;                                   // N floats
    size_t off = ((size_t)N + 63) & ~(size_t)63;
    float* A   = ws + off;                             // N*96 (GEMM output h)
    float* B   = A + (size_t)N * FEAT;                 // N*96 (layer output)
    float* out = (float*)d_out;

    const int T = 256;
    const int gbN  = (N + T - 1) / T;
    const int gbE  = (E + T - 1) / T;
    const int gbEF = (int)(((size_t)E * 24 + T - 1) / T);
    const int gbG  = (N + 15) / 16;

    // degree -> dis (once, shared by all 3 layers)
    gcn_deg_init    <<<gbN, T, 0, stream>>>(dis, N);
    gcn_deg_accum   <<<gbE, T, 0, stream>>>(dst, ew, dis, E);
    gcn_deg_finalize<<<gbN, T, 0, stream>>>(dis, N);

    // layer 1: x -> (A, B=self+bias) -> B
    gcn_gemm_wmma   <<<gbG,  192, 0, stream>>>(x, W1, dis, b1, A, B, N, 0);
    gcn_edge_scatter<<<gbEF, T,   0, stream>>>(src, dst, ew, dis, A, B, E);

    // layer 2: relu(B) -> (A, B) -> B
    gcn_gemm_wmma   <<<gbG,  192, 0, stream>>>(B, W2, dis, b2, A, B, N, 1);
    gcn_edge_scatter<<<gbEF, T,   0, stream>>>(src, dst, ew, dis, A, B, E);

    // layer 3: relu(B) -> (A, d_out) -> d_out
    gcn_gemm_wmma   <<<gbG,  192, 0, stream>>>(B, W3, dis, b3, A, out, N, 1);
    gcn_edge_scatter<<<gbEF, T,   0, stream>>>(src, dst, ew, dis, A, out, E);
}